// CapsuleLayer_23424751633174
// MI455X (gfx1250) — compile-verified
//
#include <hip/hip_runtime.h>

// Capsule dynamic routing, fused, fp32, CDNA5 WMMA (V_WMMA_F32_16X16X4_F32).
// Shapes: x[256,1152,8], W[10,1152,16,8] -> v[256,10,16]

#define B_TOT 256
#define ICAPS 1152
#define DIN   8
#define NCAP  10
#define DOUT  16
#define BT    16                 // batch tile (WMMA M/N)
#define IC    32                 // input-capsule chunk per block
#define NCHUNK (ICAPS / IC)      // 36
#define NBT    (B_TOT / BT)      // 16

typedef __attribute__((ext_vector_type(2))) float v2f;
typedef __attribute__((ext_vector_type(8))) float v8f;

__device__ __forceinline__ v8f wmma_f32_k4(v2f a, v2f b, v8f c) {
  // (neg_a, A, neg_b, B, c_mod, C, reuse_a, reuse_b)
  return __builtin_amdgcn_wmma_f32_16x16x4_f32(false, a, false, b, (short)0, c,
                                               false, false);
}

// ---------------------------------------------------------------------------
// s-pass, iteration 0: coupling is uniform 0.1 -> no softmax, no cs LDS,
// A operand comes straight from xs (aligned v2f), scale folded in at the end.
// grid = (NCHUNK, NBT), block = (32, NCAP); wave y owns capsule n=y.
// ---------------------------------------------------------------------------
__global__ __launch_bounds__(320)
void caps_route_s0(const float* __restrict__ x, const float* __restrict__ W,
                   float* __restrict__ spart)       // [chunk][b][n][j]
{
  __shared__ float xs[IC][BT][DIN];   // 16 KB

  const int lane = threadIdx.x;
  const int n    = threadIdx.y;
  const int tid  = n * 32 + lane;
  const int i0   = blockIdx.x * IC;
  const int b0   = blockIdx.y * BT;

  for (int t = tid; t < BT * IC * DIN / 4; t += 320) {
    const int e  = t * 4;
    const int d  = e & (DIN - 1);
    const int il = (e / DIN) & (IC - 1);
    const int bl = e / (DIN * IC);
    const float4 val =
        *(const float4*)(x + ((size_t)(b0 + bl) * ICAPS + (i0 + il)) * DIN + d);
    *(float4*)&xs[il][bl][d] = val;
  }
  __syncthreads();

  // A: lane -> row M=b=lane&15, K-pair by lane>>4 ; B: lane -> col N=j=lane&15
  const int bl = lane & 15;
  const int hi = lane >> 4;
  const int d0 = 2 * hi;
  const float* Wn =
      W + (size_t)n * ICAPS * DOUT * DIN + (size_t)bl * DIN;  // j = bl

  v8f acc0 = {};
  v8f acc1 = {};
  for (int il = 0; il < IC; ++il) {
    const float* wrow = Wn + (size_t)(i0 + il) * (DOUT * DIN);
    const v2f a0 = *(const v2f*)&xs[il][bl][d0];       // d0 even -> aligned
    const v2f b0v = *(const v2f*)(wrow + d0);
    acc0 = wmma_f32_k4(a0, b0v, acc0);
    const v2f a1 = *(const v2f*)&xs[il][bl][4 + d0];
    const v2f b1v = *(const v2f*)(wrow + 4 + d0);
    acc1 = wmma_f32_k4(a1, b1v, acc1);
  }

  // C layout: VGPR rr, lanes<16 -> M=rr, lanes>=16 -> M=rr+8; N=lane&15
#pragma unroll
  for (int rr = 0; rr < 8; ++rr) {
    const int brow = rr + 8 * hi;
    spart[(((size_t)blockIdx.x * B_TOT + (b0 + brow)) * NCAP + n) * DOUT + bl] =
        (acc0[rr] + acc1[rr]) * 0.1f;                  // uniform coupling 1/10
  }
}

// ---------------------------------------------------------------------------
// s-pass, iterations >0: softmax coupling over n, then same WMMA GEMM with
// A = c * x.  No branches in the inner loop.
// ---------------------------------------------------------------------------
__global__ __launch_bounds__(320)
void caps_route_s(const float* __restrict__ x, const float* __restrict__ W,
                  const float* __restrict__ blog,   // [b][i][n]
                  float* __restrict__ spart)        // [chunk][b][n][j]
{
  __shared__ float xs[IC][BT][DIN];   // 16 KB
  __shared__ float cs[NCAP][IC][BT];  // 20 KB

  const int lane = threadIdx.x;
  const int n    = threadIdx.y;
  const int tid  = n * 32 + lane;
  const int i0   = blockIdx.x * IC;
  const int b0   = blockIdx.y * BT;

  for (int t = tid; t < BT * IC * DIN / 4; t += 320) {
    const int e  = t * 4;
    const int d  = e & (DIN - 1);
    const int il = (e / DIN) & (IC - 1);
    const int bl = e / (DIN * IC);
    const float4 val =
        *(const float4*)(x + ((size_t)(b0 + bl) * ICAPS + (i0 + il)) * DIN + d);
    *(float4*)&xs[il][bl][d] = val;
  }

  // coupling coefficients: softmax over n (10 contiguous logits per (b,i))
  for (int p = tid; p < BT * IC; p += 320) {
    const int bl = p / IC;
    const int il = p - bl * IC;
    const float* bp = blog + ((size_t)(b0 + bl) * ICAPS + (i0 + il)) * NCAP;
    float e[NCAP];
    float m = -3.0e38f;
#pragma unroll
    for (int k = 0; k < NCAP; ++k) { e[k] = bp[k]; m = fmaxf(m, e[k]); }
    float sum = 0.0f;
#pragma unroll
    for (int k = 0; k < NCAP; ++k) { e[k] = __expf(e[k] - m); sum += e[k]; }
    const float inv = 1.0f / sum;
#pragma unroll
    for (int k = 0; k < NCAP; ++k) cs[k][il][bl] = e[k] * inv;
  }
  __syncthreads();

  const int bl = lane & 15;
  const int hi = lane >> 4;
  const int d0 = 2 * hi;
  const float* Wn =
      W + (size_t)n * ICAPS * DOUT * DIN + (size_t)bl * DIN;  // j = bl

  v8f acc0 = {};
  v8f acc1 = {};
  for (int il = 0; il < IC; ++il) {
    const float c = cs[n][il][bl];
    const float* wrow = Wn + (size_t)(i0 + il) * (DOUT * DIN);
    v2f a0; a0.x = c * xs[il][bl][d0];     a0.y = c * xs[il][bl][d0 + 1];
    const v2f b0v = *(const v2f*)(wrow + d0);
    acc0 = wmma_f32_k4(a0, b0v, acc0);
    v2f a1; a1.x = c * xs[il][bl][4 + d0]; a1.y = c * xs[il][bl][5 + d0];
    const v2f b1v = *(const v2f*)(wrow + 4 + d0);
    acc1 = wmma_f32_k4(a1, b1v, acc1);
  }

#pragma unroll
  for (int rr = 0; rr < 8; ++rr) {
    const int brow = rr + 8 * hi;
    spart[(((size_t)blockIdx.x * B_TOT + (b0 + brow)) * NCAP + n) * DOUT + bl] =
        acc0[rr] + acc1[rr];
  }
}

// ---------------------------------------------------------------------------
// squash: deterministically reduce the NCHUNK partial s slabs, apply squash.
// one thread per (b,n).
// ---------------------------------------------------------------------------
__global__ __launch_bounds__(256)
void caps_squash(const float* __restrict__ spart, float* __restrict__ vout)
{
  const int p = blockIdx.x * blockDim.x + threadIdx.x;  // b*NCAP + n
  if (p >= B_TOT * NCAP) return;
  float s[DOUT];
#pragma unroll
  for (int j = 0; j < DOUT; ++j) s[j] = 0.0f;
  for (int ci = 0; ci < NCHUNK; ++ci) {
    const float4* sp =
        (const float4*)(spart + ((size_t)ci * B_TOT * NCAP + p) * DOUT);
#pragma unroll
    for (int q = 0; q < 4; ++q) {
      const float4 t = sp[q];
      s[4 * q + 0] += t.x; s[4 * q + 1] += t.y;
      s[4 * q + 2] += t.z; s[4 * q + 3] += t.w;
    }
  }
  float sq = 0.0f;
#pragma unroll
  for (int j = 0; j < DOUT; ++j) sq += s[j] * s[j];
  const float scale = (sq / (1.0f + sq)) * rsqrtf(sq + 1e-7f);
  float4* o = (float4*)(vout + (size_t)p * DOUT);
#pragma unroll
  for (int q = 0; q < 4; ++q) {
    float4 t;
    t.x = s[4 * q + 0] * scale; t.y = s[4 * q + 1] * scale;
    t.z = s[4 * q + 2] * scale; t.w = s[4 * q + 3] * scale;
    o[q] = t;
  }
}

// ---------------------------------------------------------------------------
// agreement-pass: b_logits[b,i,n] (+)= sum_j u_hat[b,n,i,j]*v[b,n,j]
//   t[(i,d),b] = sum_j W[n,(i,d),j] * v[b,n,j]   (WMMA, M=(i,d) K=j N=b)
//   agreement  = sum_d x[b,i,d] * t[(i,d),b]     (per-lane 8-wide dot)
// ---------------------------------------------------------------------------
__global__ __launch_bounds__(320)
void caps_route_agr(const float* __restrict__ x, const float* __restrict__ W,
                    const float* __restrict__ v,     // [b][n][j]
                    float* __restrict__ blog,        // [b][i][n]
                    int add)
{
  __shared__ float xs[IC][BT][DIN];        // 16 KB
  __shared__ float vls[NCAP][DOUT][BT];    // 10 KB

  const int lane = threadIdx.x;
  const int n    = threadIdx.y;
  const int tid  = n * 32 + lane;
  const int i0   = blockIdx.x * IC;
  const int b0   = blockIdx.y * BT;

  for (int t = tid; t < BT * IC * DIN / 4; t += 320) {
    const int e  = t * 4;
    const int d  = e & (DIN - 1);
    const int il = (e / DIN) & (IC - 1);
    const int bl = e / (DIN * IC);
    const float4 val =
        *(const float4*)(x + ((size_t)(b0 + bl) * ICAPS + (i0 + il)) * DIN + d);
    *(float4*)&xs[il][bl][d] = val;
  }
  for (int t = tid; t < NCAP * DOUT * BT; t += 320) {
    const int b  = t & 15;
    const int j  = (t >> 4) & 15;
    const int nn = t >> 8;
    vls[nn][j][b] = v[((size_t)(b0 + b) * NCAP + nn) * DOUT + j];
  }
  __syncthreads();

  const int bl = lane & 15;
  const int hi = lane >> 4;
  // A rows: M = bl -> (i offset, d)
  const int dA = bl & 7;
  const int iA = bl >> 3;
  const float* Wn = W + (size_t)n * ICAPS * DOUT * DIN;

  for (int it = 0; it < IC / 2; ++it) {
    const int ib = i0 + 2 * it;
    const float* wA = Wn + (size_t)(ib + iA) * (DOUT * DIN) + dA;
    v8f acc = {};
#pragma unroll
    for (int kk = 0; kk < DOUT; kk += 4) {
      const int j = kk + 2 * hi;                  // lane's K-pair
      v2f a;  a.x  = wA[j * DIN];     a.y  = wA[(j + 1) * DIN];
      v2f bb; bb.x = vls[n][j][bl];   bb.y = vls[n][j + 1][bl];
      acc = wmma_f32_k4(a, bb, acc);
    }
    // lane holds t[d=0..7] for i = ib+hi, b = bl
    const int ilx = 2 * it + hi;
    float ag = 0.0f;
#pragma unroll
    for (int rr = 0; rr < 8; ++rr) ag += acc[rr] * xs[ilx][bl][rr];
    const size_t off = ((size_t)(b0 + bl) * ICAPS + (i0 + ilx)) * NCAP + n;
    blog[off] = add ? (blog[off] + ag) : ag;
  }
}

// ---------------------------------------------------------------------------
extern "C" void kernel_launch(void* const* d_in, const int* in_sizes, int n_in,
                              void* d_out, int out_size, void* d_ws,
                              size_t ws_size, hipStream_t stream)
{
  (void)in_sizes; (void)n_in; (void)out_size; (void)ws_size;
  const float* x = (const float*)d_in[0];
  const float* W = (const float*)d_in[1];
  float* out = (float*)d_out;

  // workspace layout (~18 MB):
  float* blog  = (float*)d_ws;                                   // [b][i][n]
  float* spart = blog + (size_t)B_TOT * ICAPS * NCAP;            // [ci][b][n][j]
  float* vbuf  = spart + (size_t)NCHUNK * B_TOT * NCAP * DOUT;   // [b][n][j]

  const dim3 grid(NCHUNK, NBT);
  const dim3 block(32, NCAP);
  const dim3 sgrid((B_TOT * NCAP + 255) / 256);
  const dim3 sblock(256);

  // r = 0 (uniform coupling), 1, 2; agreement updates after r=0,1
  caps_route_s0 <<<grid,  block,  0, stream>>>(x, W, spart);
  caps_squash   <<<sgrid, sblock, 0, stream>>>(spart, vbuf);
  caps_route_agr<<<grid,  block,  0, stream>>>(x, W, vbuf, blog, 0);
  caps_route_s  <<<grid,  block,  0, stream>>>(x, W, blog, spart);
  caps_squash   <<<sgrid, sblock, 0, stream>>>(spart, vbuf);
  caps_route_agr<<<grid,  block,  0, stream>>>(x, W, vbuf, blog, 1);
  caps_route_s  <<<grid,  block,  0, stream>>>(x, W, blog, spart);
  caps_squash   <<<sgrid, sblock, 0, stream>>>(spart, out);
}